// GruDirection2d_3650722201913
// MI455X (gfx1250) — compile-verified
//
#include <hip/hip_runtime.h>
#include <stdint.h>

// GRU forward-H scan: h[t] = h[t-1] + z[t]*(h~[t] - h[t-1])
// B=4, C=64, H=512, W=512  (float32 everywhere)
// Memory-bound streaming scan -> async global->LDS triple-buffer pipeline.

#define BDIM     256
#define GRU_W    512
#define GRU_H    512
#define T_CHUNK  8
#define NBUF     3
#define CHUNK_ELEMS (T_CHUNK * GRU_W)   // 4096 floats = 16 KB per tensor per chunk
#define NCHUNK   (GRU_H / T_CHUNK)      // 64

// ---- CDNA5 async global -> LDS copy (ASYNCcnt-tracked) ----
// Inline asm form of GLOBAL_LOAD_ASYNC_TO_LDS_B128 (GV addressing mode):
//   vdst = VGPR holding LDS byte address, vaddr = 64-bit global address, no saddr.
// Bypasses the clang builtin's cuda_device-address-space parameter typing.
__device__ __forceinline__ void async_copy_b128(const float* gsrc, float* ldst) {
  const uint32_t lds_addr = (uint32_t)(uintptr_t)ldst;  // low 32 bits of generic = LDS offset
  const uint64_t gaddr    = (uint64_t)(uintptr_t)gsrc;
  asm volatile("global_load_async_to_lds_b128 %0, %1, off"
               :: "v"(lds_addr), "v"(gaddr)
               : "memory");
}

template <int N>
__device__ __forceinline__ void wait_asynccnt_le() {
#if __has_builtin(__builtin_amdgcn_s_wait_asynccnt)
  __builtin_amdgcn_s_wait_asynccnt(N);
#else
  asm volatile("s_wait_asynccnt %0" :: "i"(N) : "memory");
#endif
}

__global__ __launch_bounds__(BDIM) void gru_scan_h_kernel(
    const float* __restrict__ z,
    const float* __restrict__ h_tilde,
    const float* __restrict__ h0,
    float* __restrict__ out) {
  // Triple-buffered chunk staging: 3 * (16KB + 16KB) = 96 KB LDS
  __shared__ float s_z[NBUF][CHUNK_ELEMS];
  __shared__ float s_h[NBUF][CHUNK_ELEMS];

  const int    tid   = threadIdx.x;          // 0..255
  const int    bc    = blockIdx.x;           // 0..B*C-1  (one (b,c) plane per block)
  const size_t plane = (size_t)bc * GRU_H * GRU_W;
  const int    w0    = tid * 2;              // each thread owns 2 adjacent columns

  // Issue one chunk's async copies: chunk is contiguous (rows contiguous in [H][W])
  auto issue_chunk = [&](int c, int buf) {
    const float* zc = z       + plane + (size_t)c * CHUNK_ELEMS;
    const float* hc = h_tilde + plane + (size_t)c * CHUNK_ELEMS;
#pragma unroll
    for (int r = 0; r < CHUNK_ELEMS / (BDIM * 4); ++r) {   // 4 rounds of b128
      const int flat = r * (BDIM * 4) + tid * 4;
      async_copy_b128(zc + flat, &s_z[buf][flat]);
      async_copy_b128(hc + flat, &s_h[buf][flat]);
    }
  };

  // Carry registers (two independent scan chains per thread for ILP)
  float hA = h0[(size_t)bc * GRU_W + w0];
  float hB = h0[(size_t)bc * GRU_W + w0 + 1];

  // Pipeline prologue: 2 chunks in flight (8 asyncs each per thread)
  issue_chunk(0, 0);
  issue_chunk(1, 1);

  for (int c = 0; c < NCHUNK; ++c) {
    // Keep pipeline 2 chunks deep; buffer (c+2)%3 was last read at iteration
    // c-1, whose trailing barrier already retired all readers.
    if (c + 2 < NCHUNK) issue_chunk(c + 2, (c + 2) % NBUF);

    // Async loads complete in order per wave: waiting down to the count of the
    // *younger* chunks' ops guarantees chunk c has landed in LDS.
    if (c + 2 < NCHUNK)      wait_asynccnt_le<16>();
    else if (c + 1 < NCHUNK) wait_asynccnt_le<8>();
    else                     wait_asynccnt_le<0>();
    __syncthreads();  // make all waves' async LDS writes visible

    const int buf = c % NBUF;
    float* outc = out + plane + (size_t)c * CHUNK_ELEMS;
#pragma unroll
    for (int t = 0; t < T_CHUNK; ++t) {
      const float zA = s_z[buf][t * GRU_W + w0];
      const float zB = s_z[buf][t * GRU_W + w0 + 1];
      const float cA = s_h[buf][t * GRU_W + w0];
      const float cB = s_h[buf][t * GRU_W + w0 + 1];
      // h = z*h~ + (1-z)*h  ==  h + z*(h~ - h)
      hA = __builtin_fmaf(zA, cA - hA, hA);
      hB = __builtin_fmaf(zB, cB - hB, hB);
      float2 o; o.x = hA; o.y = hB;
      *reinterpret_cast<float2*>(outc + t * GRU_W + w0) = o;  // coalesced b64 store
    }
    __syncthreads();  // protect buffer reuse at distance NBUF
  }
}

extern "C" void kernel_launch(void* const* d_in, const int* in_sizes, int n_in,
                              void* d_out, int out_size, void* d_ws, size_t ws_size,
                              hipStream_t stream) {
  const float* z  = (const float*)d_in[0];   // [B,C,H,W]
  const float* ht = (const float*)d_in[1];   // [B,C,H,W]
  const float* h0 = (const float*)d_in[2];   // [B,C,1,W]
  float*       out = (float*)d_out;          // [B,C,H,W]

  const int planes = in_sizes[0] / (GRU_H * GRU_W);  // B*C = 256
  gru_scan_h_kernel<<<planes, BDIM, 0, stream>>>(z, ht, h0, out);
}